// QuantumEntanglingLinear_43215960932474
// MI455X (gfx1250) — compile-verified
//
#include <hip/hip_runtime.h>
#include <math.h>

typedef __attribute__((ext_vector_type(4))) float v4f;

#define DDIM 1024
#define WAVES 8
#define TPB (WAVES * 32)
#define LSTRIDE 36                 // 32 data floats + 4 pad per lane chunk
#define RBUF_FLOATS (32 * LSTRIDE) // 1152 floats = 4608 B per row buffer

__device__ __forceinline__ uint32_t lds_addr(const void* p) {
  // low 32 bits of a flat pointer to LDS == LDS byte offset
  return (uint32_t)(uintptr_t)p;
}

__device__ __forceinline__ void async_b128(uint32_t lds_byte, const float* g) {
  // CDNA5 async DMA: per-lane 16B global -> LDS, tracked by ASYNCcnt
  asm volatile("global_load_async_to_lds_b128 %0, %1, off"
               :: "v"(lds_byte), "v"(g) : "memory");
}

__global__ __launch_bounds__(TPB) void qent_kernel(
    const float* __restrict__ x, const float* __restrict__ la,
    const float* __restrict__ ea, float* __restrict__ out,
    int B, int totalWaves) {
  __shared__ float2 lcs[2 * 1056];          // [layer][chunk 0..31][33] padded (c,s)
  __shared__ float2 ecs[2 * 544];           // [layer][chunk 0..31][17] padded (c,s)
  __shared__ float  rbuf[WAVES][2][RBUF_FLOATS];

  const int tid = threadIdx.x;
  // Build sin/cos tables once per block (3072 sincos, amortized over 8 rows/wave).
  for (int e = tid; e < 2 * DDIM; e += TPB) {
    int layer = e >> 10, i = e & 1023;
    float s, c;
    sincosf(la[layer * DDIM + i], &s, &c);
    lcs[layer * 1056 + (i >> 5) * 33 + (i & 31)] = make_float2(c, s);
  }
  for (int e = tid; e < DDIM; e += TPB) {
    int layer = e >> 9, i = e & 511;
    float s, c;
    sincosf(ea[layer * (DDIM / 2) + i], &s, &c);
    ecs[layer * 544 + (i >> 4) * 17 + (i & 15)] = make_float2(c, s);
  }
  __syncthreads();

  const int lane = tid & 31;
  const int wv = tid >> 5;
  const int gw = blockIdx.x * WAVES + wv;

  const float* rd0 = &rbuf[wv][0][0] + lane * LSTRIDE;
  // Async-write base: instruction k copies global chunk (32k+lane) into
  // padded lane region (4k + lane/8), slot (lane%8) -> lane-transpose for free.
  const uint32_t wb0 = lds_addr(&rbuf[wv][0][0]) +
                       (uint32_t)((lane >> 3) * (LSTRIDE * 4) + (lane & 7) * 16);

  int r = gw, cur = 0;
  if (r < B) {
    const float* gp = x + (size_t)r * DDIM + lane * 4;
#pragma unroll
    for (int k = 0; k < 8; ++k) async_b128(wb0 + k * (4 * LSTRIDE * 4), gp + k * 128);
  }

  while (r < B) {
    const int rn = r + totalWaves;
    if (rn < B) {
      const float* gp = x + (size_t)rn * DDIM + lane * 4;
      const uint32_t wb = wb0 + (uint32_t)((cur ^ 1) * (RBUF_FLOATS * 4));
#pragma unroll
      for (int k = 0; k < 8; ++k) async_b128(wb + k * (4 * LSTRIDE * 4), gp + k * 128);
      asm volatile("s_wait_asynccnt 0x8" ::: "memory"); // current row's 8 DMAs done
    } else {
      asm volatile("s_wait_asynccnt 0x0" ::: "memory");
    }

    // Pull this lane's 32 contiguous row elements from LDS.
    const float* bp = rd0 + cur * RBUF_FLOATS;
    float xv[32];
#pragma unroll
    for (int k = 0; k < 8; ++k) {
      v4f v = *(const v4f*)(bp + 4 * k);
      xv[4 * k + 0] = v.x; xv[4 * k + 1] = v.y;
      xv[4 * k + 2] = v.z; xv[4 * k + 3] = v.w;
    }

#pragma unroll
    for (int layer = 0; layer < 2; ++layer) {
      const float2* lt = &lcs[layer * 1056 + lane * 33];
      float xn0 = __shfl_down(xv[0], 1); // next lane's first element (lane31: unused garbage)

      // Phase A: per-lane affine reduce of t-recurrence: t_last = alpha*T + beta
      float alpha = 1.0f, beta = 0.0f;
#pragma unroll
      for (int k = 1; k < 32; ++k) {
        float2 cs = lt[k - 1];
        alpha = cs.y * alpha;
        beta = fmaf(cs.y, beta, cs.x * xv[k]);
      }
      float2 cs31 = lt[31];
      float A = cs31.y * alpha;
      float Bv = fmaf(cs31.y, beta, cs31.x * xn0);

      // Wave-level inclusive scan of affine maps (Hillis-Steele, 5 steps)
#pragma unroll
      for (int d = 1; d < 32; d <<= 1) {
        float Ap = __shfl_up(A, d);
        float Bp = __shfl_up(Bv, d);
        if (lane >= d) { Bv = fmaf(A, Bp, Bv); A = A * Ap; }
      }
      float Apre = __shfl_up(A, 1);
      float Bpre = __shfl_up(Bv, 1);
      float T0 = __shfl(xv[0], 0);
      float T = (lane == 0) ? T0 : fmaf(Apre, T0, Bpre);

      // Phase B: replay segment, emit local-layer outputs
      float o[32];
      float t = T, tD1 = 0.0f;
#pragma unroll
      for (int k = 0; k < 32; ++k) {
        float2 cs = lt[k];
        float xn = (k < 31) ? xv[k + 1] : xn0;
        if (k == 31) tD1 = t;                 // t_{base+31}; lane31: t_{D-1}
        o[k] = fmaf(cs.x, t, -cs.y * xn);
        t = fmaf(cs.y, t, cs.x * xn);
      }
      // Wrap-around step (i = D-1 rotates with provisional out_0)
      float w0 = __shfl(o[0], 0);
      float tLast = __shfl(tD1, 31);
      float2 csl = lcs[layer * 1056 + 31 * 33 + 31]; // (c,s) of angle D-1
      if (lane == 31) o[31] = fmaf(csl.x, tLast, -csl.y * w0);
      if (lane == 0)  o[0]  = fmaf(csl.y, tLast,  csl.x * w0);

      // Entangler layer: disjoint pairs, fully lane-local
      const float2* et = &ecs[layer * 544 + lane * 17];
#pragma unroll
      for (int m = 0; m < 16; ++m) {
        float2 ce = et[m];
        float a = o[2 * m], b = o[2 * m + 1];
        xv[2 * m]     = fmaf(ce.x, a, -ce.y * b);
        xv[2 * m + 1] = fmaf(ce.y, a,  ce.x * b);
      }
    }

    // Non-temporal b128 stores straight from registers.
    float* op = out + (size_t)r * DDIM + lane * 32;
#pragma unroll
    for (int k = 0; k < 8; ++k) {
      v4f v = { xv[4 * k], xv[4 * k + 1], xv[4 * k + 2], xv[4 * k + 3] };
      __builtin_nontemporal_store(v, (v4f*)(op + 4 * k));
    }

    cur ^= 1;
    r = rn;
  }
}

extern "C" void kernel_launch(void* const* d_in, const int* in_sizes, int n_in,
                              void* d_out, int out_size, void* d_ws, size_t ws_size,
                              hipStream_t stream) {
  const float* x  = (const float*)d_in[0];
  const float* la = (const float*)d_in[1];
  const float* ea = (const float*)d_in[2];
  float* out = (float*)d_out;
  const int B = in_sizes[0] / DDIM;   // 32768 rows
  const int blocks = 512;             // 512 x 8 waves = 4096 waves -> 8 rows per wave
  qent_kernel<<<blocks, TPB, 0, stream>>>(x, la, ea, out, B, blocks * WAVES);
}